// PeakDetection_9105330667526
// MI455X (gfx1250) — compile-verified
//
#include <hip/hip_runtime.h>
#include <hip/hip_bf16.h>
#include <stdint.h>

// Problem constants (reference: inputs (16,1024,1024,3) fp32, threshold 0.995,
// diamond radius-2 NMS, nonzero(size=65536, fill=-1))
#define BATCH     16
#define HEIGHT    1024
#define WIDTH     1024
#define CHANS     3
#define ROWLEN    (WIDTH * CHANS)        // 3072 floats per image row
#define NROWS     (BATCH * HEIGHT)       // 16384 image rows
#define NTHR      1024                   // 32 wave32 waves per block
#define WPR       32                     // ballot words (waves) per image row
#define MAX_PEAKS 65536
#define NEGINF    (-3.4028234663852886e38f)
#define THRESH    0.995f

// Tile geometry for the mask kernel
#define TW        256                    // w extent per tile
#define TH        8                      // h extent per tile
#define LROWF     (TW * CHANS + 16)      // 784 floats: 8-float pad both sides
#define LNROW     (TH + 4)               // 12 staged rows (2 halo each side)
#define CHUNKS    (LROWF / 4)            // 196 float4 chunks per staged row
#define WTILES    (WIDTH / TW)           // 4
#define HTILES    (HEIGHT / TH)          // 128
#define NBLOCKS   (BATCH * HTILES * WTILES) // 8192

// ---------------------------------------------------------------------------
// CDNA5 async global->LDS DMA (ASYNCcnt-tracked), per-lane 128-bit transfers.
// vdst = 32-bit wave-relative LDS byte address, vaddr = 64-bit global address.
// ---------------------------------------------------------------------------
__device__ __forceinline__ void async_load_b128(uint32_t lds_byte_addr,
                                                const float* gaddr) {
  asm volatile("global_load_async_to_lds_b128 %0, %1, off"
               :: "v"(lds_byte_addr), "v"(gaddr)
               : "memory");
}

__device__ __forceinline__ void wait_asynccnt0() {
  asm volatile("s_wait_asynccnt 0" ::: "memory");
}

// ---------------------------------------------------------------------------
// Kernel 0a/0b: initialize output (-1 pad) and per-row counts (0).
// ---------------------------------------------------------------------------
__global__ void fill_neg1_kernel(int* __restrict__ out, int n) {
  int i = blockIdx.x * blockDim.x + threadIdx.x;
  if (i < n) out[i] = -1;
}
__global__ void zero_kernel(int* __restrict__ p, int n) {
  int i = blockIdx.x * blockDim.x + threadIdx.x;
  if (i < n) p[i] = 0;
}

// ---------------------------------------------------------------------------
// Kernel 1: tiled peak mask. One block = 256(w) x 8(h) tile of one image.
// Stages a 12 x 784-float halo window via async DMA (all boundary chunks are
// whole float4s thanks to the 8-float side pads -> 16B-aligned B128 loads),
// computes the diamond-NMS mask branch-free from LDS, emits wave32 ballots
// to ballots[row][w/32][c] and atomically accumulates per-row counts.
// ---------------------------------------------------------------------------
__global__ __launch_bounds__(NTHR) void peak_mask_kernel(
    const float* __restrict__ in,
    unsigned*    __restrict__ ballots,   // [NROWS][WPR][3]
    int*         __restrict__ counts)    // [NROWS]
{
  __shared__ __align__(16) float tile[LNROW * LROWF]; // 37,632 B
  __shared__ int cnt[TH][TW / 32];                    // [8][8]

  const int t  = threadIdx.x;
  const int bx = blockIdx.x;
  const int wt = bx & (WTILES - 1);                   // w tile
  const int ht = (bx >> 2) & (HTILES - 1);            // h tile
  const int b  = bx >> 9;                             // batch
  const int w0 = wt * TW;
  const int h0 = ht * TH;
  const float* img = in + (size_t)b * HEIGHT * ROWLEN;

  // ---- stage 12 rows x 784 floats; NEG_INF outside the image -------------
  for (int idx = t; idx < LNROW * CHUNKS; idx += NTHR) {
    const int r   = idx / CHUNKS;                     // staged row 0..11
    const int ck  = idx - r * CHUNKS;                 // chunk in row
    const int src_h = h0 + r - 2;
    const int g0  = w0 * CHANS - 8 + ck * 4;          // float idx in source row
    float* ldst = &tile[r * LROWF + ck * 4];
    if (src_h >= 0 && src_h < HEIGHT && g0 >= 0 && g0 <= ROWLEN - 4) {
      async_load_b128((uint32_t)(size_t)ldst,
                      img + (size_t)src_h * ROWLEN + g0);
    } else {
      *(float4*)ldst = make_float4(NEGINF, NEGINF, NEGINF, NEGINF);
    }
  }
  wait_asynccnt0();
  __syncthreads();

  // ---- diamond (Manhattan radius 2) max, branch-free from staged halo ----
  const int w_off = t & (TW - 1);                     // 0..255
  const int lane  = t & 31;
  const int wir   = (t >> 5) & 7;                     // wave within row
#pragma unroll
  for (int i = 0; i < 2; ++i) {
    const int hh = (t >> 8) + 4 * i;                  // 0..7
    const int r  = hh + 2;                            // center staged row
    unsigned mybits = 0;
#pragma unroll
    for (int c = 0; c < CHANS; ++c) {
      const int col = 8 + w_off * CHANS + c;
      const float* ctr = &tile[r * LROWF + col];
      const float x = ctr[0];
      float m;
      m =          ctr[-2 * LROWF];                               // (-2, 0)
      m = fmaxf(m, ctr[-LROWF - CHANS]);                          // (-1,-1)
      m = fmaxf(m, ctr[-LROWF]);                                  // (-1, 0)
      m = fmaxf(m, ctr[-LROWF + CHANS]);                          // (-1, 1)
      m = fmaxf(m, ctr[-2 * CHANS]);                              // ( 0,-2)
      m = fmaxf(m, ctr[-CHANS]);                                  // ( 0,-1)
      m = fmaxf(m, ctr[CHANS]);                                   // ( 0, 1)
      m = fmaxf(m, ctr[2 * CHANS]);                               // ( 0, 2)
      m = fmaxf(m, ctr[LROWF - CHANS]);                           // ( 1,-1)
      m = fmaxf(m, ctr[LROWF]);                                   // ( 1, 0)
      m = fmaxf(m, ctr[LROWF + CHANS]);                           // ( 1, 1)
      m = fmaxf(m, ctr[2 * LROWF]);                               // ( 2, 0)
      const bool peak = (x >= m) && (x > THRESH);
      mybits |= (unsigned)peak << c;
    }
    const unsigned bl0 = __builtin_amdgcn_ballot_w32((mybits >> 0) & 1);
    const unsigned bl1 = __builtin_amdgcn_ballot_w32((mybits >> 1) & 1);
    const unsigned bl2 = __builtin_amdgcn_ballot_w32((mybits >> 2) & 1);
    if (lane == 0) {
      const int row = b * HEIGHT + h0 + hh;
      unsigned* bb = ballots + ((size_t)row * WPR + wt * 8 + wir) * 3;
      bb[0] = bl0; bb[1] = bl1; bb[2] = bl2;
      cnt[hh][wir] = __popc(bl0) + __popc(bl1) + __popc(bl2);
    }
  }
  __syncthreads();
  if (t < TH) {
    int s = 0;
#pragma unroll
    for (int j = 0; j < TW / 32; ++j) s += cnt[t][j];
    atomicAdd(&counts[b * HEIGHT + h0 + t], s);
  }
}

// ---------------------------------------------------------------------------
// Kernel 2: exclusive prefix scan of 16384 row counts (single block).
// ---------------------------------------------------------------------------
__global__ __launch_bounds__(NTHR) void scan_kernel(
    const int* __restrict__ counts, int* __restrict__ offs)
{
  __shared__ int part[NTHR];
  const int t = threadIdx.x;
  int local[NROWS / NTHR];                 // 16 per thread
  int s = 0;
#pragma unroll
  for (int i = 0; i < NROWS / NTHR; ++i) {
    local[i] = counts[t * (NROWS / NTHR) + i];
    s += local[i];
  }
  part[t] = s;
  __syncthreads();
  for (int d = 1; d < NTHR; d <<= 1) {     // Hillis-Steele inclusive scan
    int v = (t >= d) ? part[t - d] : 0;
    __syncthreads();
    part[t] += v;
    __syncthreads();
  }
  int base = (t == 0) ? 0 : part[t - 1];
#pragma unroll
  for (int i = 0; i < NROWS / NTHR; ++i) {
    offs[t * (NROWS / NTHR) + i] = base;
    base += local[i];
  }
  if (t == NTHR - 1) offs[NROWS] = base;
}

// ---------------------------------------------------------------------------
// Kernel 3: ordered emission from ballots. Global order = row-major (b,h,w,c),
// exactly jnp.nonzero order; rank = row_base + wave_base + lane prefix.
// ---------------------------------------------------------------------------
__global__ __launch_bounds__(NTHR) void emit_kernel(
    const unsigned* __restrict__ ballots,
    const int*      __restrict__ offs,
    int*            __restrict__ out)     // [MAX_PEAKS][4]
{
  __shared__ int wbase[WPR];
  const int row     = blockIdx.x;
  const int rowbase = offs[row];
  if (rowbase >= MAX_PEAKS) return;       // uniform per block

  const int t = threadIdx.x, lane = t & 31, wave = t >> 5;
  const unsigned* bb = ballots + ((size_t)row * WPR + wave) * 3;
  const unsigned b0 = bb[0], b1 = bb[1], b2 = bb[2];
  if (lane == 0) wbase[wave] = __popc(b0) + __popc(b1) + __popc(b2);
  __syncthreads();
  if (t == 0) {
    int s = 0;
#pragma unroll
    for (int i = 0; i < WPR; ++i) { int c = wbase[i]; wbase[i] = s; s += c; }
  }
  __syncthreads();

  const unsigned below = (1u << lane) - 1u;
  int pos = rowbase + wbase[wave]
          + __popc(b0 & below) + __popc(b1 & below) + __popc(b2 & below);

  const int h = row & (HEIGHT - 1);
  const int b = row >> 10;
  const int w = t;
  const unsigned mb = ((b0 >> lane) & 1u)
                    | (((b1 >> lane) & 1u) << 1)
                    | (((b2 >> lane) & 1u) << 2);
#pragma unroll
  for (int c = 0; c < CHANS; ++c) {
    if ((mb >> c) & 1u) {
      if (pos < MAX_PEAKS) {
        *(int4*)(out + (size_t)pos * 4) = make_int4(b, h, w, c);
      }
      ++pos;
    }
  }
}

// ---------------------------------------------------------------------------
extern "C" void kernel_launch(void* const* d_in, const int* in_sizes, int n_in,
                              void* d_out, int out_size, void* d_ws, size_t ws_size,
                              hipStream_t stream) {
  const float* in = (const float*)d_in[0];
  int* out = (int*)d_out;                     // (65536, 4) int32 indices

  // workspace layout: ballots (16384*32*3 u32 = 6.29 MB), counts, offs
  unsigned* ballots = (unsigned*)d_ws;
  int* counts = (int*)(ballots + (size_t)NROWS * WPR * 3);
  int* offs   = counts + NROWS;               // NROWS+1 ints

  fill_neg1_kernel<<<(out_size + NTHR - 1) / NTHR, NTHR, 0, stream>>>(out, out_size);
  zero_kernel<<<(NROWS + NTHR - 1) / NTHR, NTHR, 0, stream>>>(counts, NROWS);
  peak_mask_kernel<<<NBLOCKS, NTHR, 0, stream>>>(in, ballots, counts);
  scan_kernel<<<1, NTHR, 0, stream>>>(counts, offs);
  emit_kernel<<<NROWS, NTHR, 0, stream>>>(ballots, offs, out);
}